// PolynomialModule_2860448219241
// MI455X (gfx1250) — compile-verified
//
#include <hip/hip_runtime.h>

typedef __attribute__((ext_vector_type(2))) float v2f;
typedef __attribute__((ext_vector_type(8))) float v8f;
typedef __attribute__((ext_vector_type(4))) unsigned int v4u;
typedef __attribute__((ext_vector_type(8))) int v8i;
typedef __attribute__((ext_vector_type(4))) int v4i;

#define B_SZ 4096
#define D_SZ 128
#define O_SZ 128
#define KREAL 129
#define KP 132              // K padded to multiple of 4 for WMMA f32 16x16x4
#define EPSV 1e-5f

// ws float layout:
// [0,128)    sum1 (per input column)
// [128,256)  sumsq1
// [256,384)  a1 = gamma1 * invstd1
// [384,512)  b1 = beta1 - mu1 * a1
// [512,640)  sum2 (per output channel)
// [640,768)  sumsq2
// [768,896)  scale2 = gamma2 * invstd2
// [896,1024) shift2 = beta2 - mu2 * scale2

__global__ void zero_ws_kernel(float* __restrict__ ws) {
  int i = blockIdx.x * blockDim.x + threadIdx.x;
  if (i < 1024) ws[i] = 0.0f;
}

// Per-column sums of x (4096 x 128). 64 blocks x 256 threads, 64 rows per block (32 per thread).
// col = (j*256 + t) & 127 == t & 127  (256 ≡ 0 mod 128) → register accumulate.
__global__ void bn1_partial_kernel(const float* __restrict__ x, float* __restrict__ ws) {
  __shared__ float s_sum[128];
  __shared__ float s_sq[128];
  const int t = threadIdx.x;
  if (t < 128) { s_sum[t] = 0.0f; s_sq[t] = 0.0f; }
  __syncthreads();
  const size_t base = (size_t)blockIdx.x * 8192;
  float ls = 0.0f, lq = 0.0f;
  for (int j = 0; j < 32; ++j) {
    float v = x[base + (size_t)j * 256 + t];
    ls += v;
    lq += v * v;
  }
  const int c = t & 127;
  atomicAdd(&s_sum[c], ls);
  atomicAdd(&s_sq[c], lq);
  __syncthreads();
  if (t < 128) {
    atomicAdd(&ws[t], s_sum[t]);
    atomicAdd(&ws[128 + t], s_sq[t]);
  }
}

__global__ void bn1_finalize_kernel(const float* __restrict__ g1,
                                    const float* __restrict__ be1,
                                    float* __restrict__ ws) {
  int t = threadIdx.x;
  if (t >= 128) return;
  float mu  = ws[t] * (1.0f / 4096.0f);
  float var = ws[128 + t] * (1.0f / 4096.0f) - mu * mu;
  float inv = rsqrtf(var + EPSV);
  float a = g1[t] * inv;
  ws[256 + t] = a;
  ws[384 + t] = be1[t] - mu * a;
}

// One workgroup (256 thr = 8 waves) per batch row.
__global__ __launch_bounds__(256)
void poly_gemm_kernel(const float* __restrict__ x,
                      const float* __restrict__ fcw,
                      const float* __restrict__ fcb,
                      float* __restrict__ out,
                      float* __restrict__ ws) {
  __shared__ float Fl[D_SZ * KP];   // relu(poly) features, K-padded
  __shared__ float Wl[O_SZ * KP];   // fc_w, K-padded (stride 132 dwords)
  __shared__ float xn[D_SZ];
  __shared__ float s2[O_SZ];
  __shared__ float q2[O_SZ];

  const int t = threadIdx.x;
  const int b = blockIdx.x;

#if __has_builtin(__builtin_amdgcn_tensor_load_to_lds)
  // --- Tensor Data Mover: fc_w (128 rows x 129 dwords) -> Wl, LDS stride 132.
  // Iterate mode: tile = 1 row of 129 dwords, 128 iterations,
  // global += 129 dwords, lds += 132 dwords per iteration.
  if (t < 32) {                       // wave 0 only (TDM ignores EXEC; one issue)
    unsigned lds_base = (unsigned)(size_t)(void*)&Wl[0];
    unsigned long long ga = (unsigned long long)(size_t)fcw;

    v4u g0;
    g0[0] = 1u;                                   // count=1, user descriptor
    g0[1] = lds_base;                             // lds_addr (bytes)
    g0[2] = (unsigned)(ga & 0xFFFFFFFFu);         // global_addr[31:0]
    g0[3] = (unsigned)((ga >> 32) & 0x01FFFFFFu)  // global_addr[56:32]
          | 0x80000000u;                          // type=2 ("image") at [127:126]

    v8i g1;
    g1[0] = (2 << 16)                             // data_size = 4 bytes
          | (1 << 19);                            // iterate_enable
    g1[1] = (int)((KREAL & 0xFFFF) << 16);        // tensor_dim0[15:0] at [63:48]
    g1[2] = (int)((KREAL >> 16) | (1u << 16));    // tensor_dim0[31:16] | tensor_dim1[15:0]
    g1[3] = (int)((unsigned)KREAL << 16);         // tensor_dim1[31:16]=0 | tile_dim0=129
    g1[4] = 1;                                    // tile_dim1=1 | tile_dim2=0
    g1[5] = KREAL;                                // tensor_dim0_stride[31:0]
    g1[6] = 0;                                    // stride0[47:32] | stride1[15:0]
    g1[7] = 0;                                    // tensor_dim1_stride[47:16]

    v4i g2;
    g2[0] = 0;                                    // tensor_dim2 (unused)
    g2[1] = KP;                                   // lds_addr_increment (dwords)
    g2[2] = KREAL;                                // global_addr_increment[31:0]
    g2[3] = (int)(127u << 16);                    // incr[47:32]=0 | iterate_count=127 (128x)

    v4i g3 = {0, 0, 0, 0};
    v8i g4 = {0, 0, 0, 0, 0, 0, 0, 0};            // extra group (clang-23 6-arg form)

    __builtin_amdgcn_tensor_load_to_lds(g0, g1, g2, g3, g4, 0);
  }
  // Zero the 3 pad columns per row (disjoint from TDM-written dwords).
  if (t < 128) {
    Wl[t * KP + 129] = 0.0f;
    Wl[t * KP + 130] = 0.0f;
    Wl[t * KP + 131] = 0.0f;
  }
#else
  for (int idx = t; idx < O_SZ * KP; idx += 256) {
    int o = idx / KP;
    int k = idx - o * KP;
    Wl[idx] = (k < KREAL) ? fcw[o * KREAL + k] : 0.0f;
  }
#endif

  if (t < 128) {
    float v = x[(size_t)b * D_SZ + t];
    xn[t] = v * ws[256 + t] + ws[384 + t];
    s2[t] = 0.0f;
    q2[t] = 0.0f;
  }
  __syncthreads();   // xn ready (TDM still in flight; overlaps feature build)

  // Build relu(poly): flat p = [xn(128), outer(xn,xn) row-major(16384)],
  // feats[d][k] = p[d*129 + k], pad k in [129,132) with 0.
  for (int idx = t; idx < D_SZ * KP; idx += 256) {
    int d = idx / KP;
    int k = idx - d * KP;
    float v = 0.0f;
    if (k < KREAL) {
      int f = d * KREAL + k;
      if (f < 128) {
        v = xn[f];
      } else {
        int q = f - 128;
        v = xn[q >> 7] * xn[q & 127];
      }
    }
    Fl[idx] = fmaxf(v, 0.0f);
  }

#if __has_builtin(__builtin_amdgcn_tensor_load_to_lds)
  if (t < 32) {
    __builtin_amdgcn_s_wait_tensorcnt(0);   // wave 0 drains its TDM op
  }
#endif
  __syncthreads();   // F and W both published to the workgroup

  // Wave w owns output columns [w*16, w*16+16); 8 row tiles of 16.
  const int wave = t >> 5;
  const int lane = t & 31;
  const int l16  = lane & 15;
  const int kofs = (lane >> 4) << 1;   // A/B frag: K = kofs + {0,1}

  v8f acc[8];
  v8f zv = {0.f, 0.f, 0.f, 0.f, 0.f, 0.f, 0.f, 0.f};
#pragma unroll
  for (int i = 0; i < 8; ++i) acc[i] = zv;

  const float* wrow = &Wl[(wave * 16 + l16) * KP + kofs];
  const float* frow = &Fl[l16 * KP + kofs];

  for (int kk = 0; kk < KP; kk += 4) {
    v2f bf = *(const v2f*)(wrow + kk);
#pragma unroll
    for (int tm = 0; tm < 8; ++tm) {
      v2f af = *(const v2f*)(frow + tm * 16 * KP + kk);
      acc[tm] = __builtin_amdgcn_wmma_f32_16x16x4_f32(
          false, af, false, bf, (short)0, acc[tm], false, false);
    }
  }

  // Epilogue: add bias, store pre-BN result, accumulate channel stats.
  const int o = wave * 16 + l16;
  const float bias = fcb[o];
  const int mhalf = (lane >> 4) * 8;   // C/D layout: M = r + 8*(lane/16)
  float ls = 0.0f, lq = 0.0f;
#pragma unroll
  for (int tm = 0; tm < 8; ++tm) {
    int mbase = tm * 16 + mhalf;
#pragma unroll
    for (int r = 0; r < 8; ++r) {
      float v = acc[tm][r] + bias;
      out[((size_t)b * D_SZ + (mbase + r)) * O_SZ + o] = v;
      ls += v;
      lq += v * v;
    }
  }
  atomicAdd(&s2[o], ls);
  atomicAdd(&q2[o], lq);
  __syncthreads();
  if (t < 128) {
    atomicAdd(&ws[512 + t], s2[t]);
    atomicAdd(&ws[640 + t], q2[t]);
  }
}

__global__ void bn2_finalize_kernel(const float* __restrict__ g2,
                                    const float* __restrict__ be2,
                                    float* __restrict__ ws) {
  int t = threadIdx.x;
  if (t >= 128) return;
  const float invn = 1.0f / (4096.0f * 128.0f);
  float mu  = ws[512 + t] * invn;
  float var = ws[640 + t] * invn - mu * mu;
  float inv = rsqrtf(var + EPSV);
  float sc = g2[t] * inv;
  ws[768 + t] = sc;
  ws[896 + t] = be2[t] - mu * sc;
}

// In-place channel-wise normalize, float4 vectorized (o stride 1, rows of 128).
__global__ void bn2_apply_kernel(float* __restrict__ out, const float* __restrict__ ws) {
  size_t i4 = (size_t)blockIdx.x * blockDim.x + threadIdx.x;
  size_t base = i4 * 4;                  // total 4096*128*128 elements
  int o0 = (int)(base & 127);
  float4* p = (float4*)(out + base);
  float4 v = *p;
  v.x = v.x * ws[768 + o0 + 0] + ws[896 + o0 + 0];
  v.y = v.y * ws[768 + o0 + 1] + ws[896 + o0 + 1];
  v.z = v.z * ws[768 + o0 + 2] + ws[896 + o0 + 2];
  v.w = v.w * ws[768 + o0 + 3] + ws[896 + o0 + 3];
  *p = v;
}

extern "C" void kernel_launch(void* const* d_in, const int* in_sizes, int n_in,
                              void* d_out, int out_size, void* d_ws, size_t ws_size,
                              hipStream_t stream) {
  const float* x   = (const float*)d_in[0];
  const float* g1  = (const float*)d_in[1];
  const float* be1 = (const float*)d_in[2];
  const float* fcw = (const float*)d_in[3];
  const float* fcb = (const float*)d_in[4];
  const float* g2  = (const float*)d_in[5];
  const float* be2 = (const float*)d_in[6];
  float* out = (float*)d_out;
  float* ws  = (float*)d_ws;

  zero_ws_kernel<<<4, 256, 0, stream>>>(ws);
  bn1_partial_kernel<<<64, 256, 0, stream>>>(x, ws);
  bn1_finalize_kernel<<<1, 128, 0, stream>>>(g1, be1, ws);
  poly_gemm_kernel<<<B_SZ, 256, 0, stream>>>(x, fcw, fcb, out, ws);
  bn2_finalize_kernel<<<1, 128, 0, stream>>>(g2, be2, ws);
  bn2_apply_kernel<<<65536, 256, 0, stream>>>(out, ws);
}